// SAM_81544249082106
// MI455X (gfx1250) — compile-verified
//
#include <hip/hip_runtime.h>
#include <hip/hip_bf16.h>

typedef __attribute__((ext_vector_type(2))) float v2f;
typedef __attribute__((ext_vector_type(8))) float v8f;

__device__ __forceinline__ v8f v8zero() {
  v8f z = {0.f, 0.f, 0.f, 0.f, 0.f, 0.f, 0.f, 0.f};
  return z;
}

// D = A(16x4 f32) * B(4x16 f32) + C(16x16 f32), native CDNA5 f32 WMMA
__device__ __forceinline__ v8f wmma4(v2f a, v2f b, v8f c) {
  return __builtin_amdgcn_wmma_f32_16x16x4_f32(
      /*neg_a=*/false, a, /*neg_b=*/false, b,
      /*c_mod=*/(short)0, c, /*reuse_a=*/false, /*reuse_b=*/false);
}

__device__ __forceinline__ float sigm(float x) { return 1.0f / (1.0f + __expf(-x)); }

// ---------------------------------------------------------------------------
// 3x3 SAME conv, Cout=32 fixed, arbitrary Cin, no bias, optional ReLU.
// GEMM view: M = N*H*W = 65536, N = 32 (two 16-wide WMMA tiles), K = Cin*9,
// reduction ordered tap-outer (9, unrolled; constant dy/dx, one branchless
// padding mask per tap) x Cin-inner (K-step 4). One wave: 16(M) x 32(N) tile.
// Block = 8 waves. Grid = 512.
// ---------------------------------------------------------------------------
__global__ __launch_bounds__(256)
void conv3x3_wmma(const float* __restrict__ in, const float* __restrict__ wgt,
                  float* __restrict__ out, int Cin, int relu) {
  const int lane  = threadIdx.x & 31;
  const int wave  = threadIdx.x >> 5;
  const int lnr   = lane & 15;
  const int khalf = (lane >> 4) << 1;  // 0 or 2: which K pair this lane feeds
  const int m0    = (blockIdx.x * 8 + wave) << 4;
  const int row   = m0 + lnr;
  const int n     = row >> 14;
  const int hw    = row & 16383;
  const int h     = hw >> 7;
  const int w     = hw & 127;
  const int K     = Cin * 9;
  const float* inN = in + (((size_t)n * Cin) << 14);

  v8f acc0 = v8zero(), acc1 = v8zero();
#pragma unroll
  for (int tap = 0; tap < 9; ++tap) {
    const int dy = tap / 3 - 1;           // compile-time after unroll
    const int dx = tap - (tap / 3) * 3 - 1;
    const int hh = h + dy, ww = w + dx;
    const bool valid = ((unsigned)hh < 128u) & ((unsigned)ww < 128u);
    const float msk = valid ? 1.0f : 0.0f;
    const int hc = valid ? hh : h;        // clamp to an always-valid pixel
    const int wc = valid ? ww : w;
    const float* pin = inN + (hc << 7) + wc;
    const float* pw0 = wgt + lnr * K + tap;
    const float* pw1 = wgt + (lnr + 16) * K + tap;
#pragma unroll 4
    for (int c0 = 0; c0 < Cin; c0 += 4) {
      const int c = c0 + khalf;
      v2f a, b0, b1;
      a[0]  = pin[(size_t)c << 14] * msk;
      a[1]  = pin[(size_t)(c + 1) << 14] * msk;
      b0[0] = pw0[c * 9];
      b0[1] = pw0[(c + 1) * 9];
      b1[0] = pw1[c * 9];
      b1[1] = pw1[(c + 1) * 9];
      acc0 = wmma4(a, b0, acc0);
      acc1 = wmma4(a, b1, acc1);
    }
  }

  const int rbase = m0 + ((lane >> 4) << 3);
#pragma unroll
  for (int r = 0; r < 8; ++r) {
    int m   = rbase + r;
    int nn  = m >> 14;
    int mhw = m & 16383;
    float v0 = acc0[r], v1 = acc1[r];
    if (relu) { v0 = fmaxf(v0, 0.0f); v1 = fmaxf(v1, 0.0f); }
    out[(((size_t)nn * 32 + lnr) << 14) + mhw]      = v0;
    out[(((size_t)nn * 32 + lnr + 16) << 14) + mhw] = v1;
  }
}

// ---------------------------------------------------------------------------
// BiLSTM scan over T=128 steps with fused depthwise-1x1 input transform and
// on-the-fly input projection. One wave owns 16 sequence rows; h,c live in
// registers (WMMA C/D layout); Wih/Whh staged in LDS; per-step gates GEMM is
// 16x128 with K=64 (32 from x_t, 32 from h) -> 128 v_wmma per step.
// h relayout (C/D form -> A fragments) goes through a wave-private LDS slice;
// same-wave LDS ops are in-order (DScnt), so no block barrier is needed.
// hs output layout: (512 seq, 128 t, 32 ch).
// vertical: seq = n*128+h, scan over w.  horizontal: seq = n*128+w, scan over h.
// ---------------------------------------------------------------------------
__global__ __launch_bounds__(128)
void lstm_scan_wmma(const float* __restrict__ xin,
                    const float* __restrict__ fmw, const float* __restrict__ fmb,
                    const float* __restrict__ Wih, const float* __restrict__ Whh,
                    const float* __restrict__ bih, const float* __restrict__ bhh,
                    float* __restrict__ hs, int vertical, int reverse) {
  __shared__ float sWih[128 * 32];
  __shared__ float sWhh[128 * 32];
  __shared__ float sh[4 * 16 * 32];

  for (int i = threadIdx.x; i < 4096; i += 128) {
    sWih[i] = Wih[i];
    sWhh[i] = Whh[i];
  }
  __syncthreads();

  const int lane  = threadIdx.x & 31;
  const int wave  = threadIdx.x >> 5;
  const int lnr   = lane & 15;
  const int hi    = lane >> 4;
  const int khalf = hi << 1;
  const int m0    = (blockIdx.x * 4 + wave) << 4;

  const int srow = m0 + lnr;
  const int n = srow >> 7, p = srow & 127;
  const size_t xbase = (((size_t)n * 32) << 14) + (vertical ? (p << 7) : p);
  const int tmul = vertical ? 1 : 128;

  float bcol[8];
#pragma unroll
  for (int j = 0; j < 8; ++j) {
    int g = j * 16 + lnr;
    bcol[j] = bih[g] + bhh[g];
  }
  float dwm[16], dwb[16];
#pragma unroll
  for (int kk = 0; kk < 8; ++kk) {
#pragma unroll
    for (int e = 0; e < 2; ++e) {
      int c = kk * 4 + khalf + e;
      dwm[kk * 2 + e] = fmw[c];
      dwb[kk * 2 + e] = fmb[c];
    }
  }

  float ha[16];
#pragma unroll
  for (int i = 0; i < 16; ++i) ha[i] = 0.0f;
  v8f cst[2];
  cst[0] = v8zero();
  cst[1] = v8zero();

  float* shw = sh + wave * 512;
  const int rbase_off = hi << 3;

  for (int step = 0; step < 128; ++step) {
    const int t = reverse ? (127 - step) : step;
    const int tn = reverse ? (t > 0 ? t - 1 : 0) : (t < 127 ? t + 1 : 127);
    __builtin_prefetch(xin + xbase + (size_t)tn * tmul, 0, 1);

    v8f acc[8];
#pragma unroll
    for (int j = 0; j < 8; ++j) acc[j] = v8zero();

    // gates += x_t @ Wih^T  (depthwise fused into the load)
#pragma unroll
    for (int kk = 0; kk < 8; ++kk) {
      const int c = kk * 4 + khalf;
      v2f a;
      a[0] = xin[xbase + ((size_t)c << 14) + (size_t)t * tmul]       * dwm[kk * 2]     + dwb[kk * 2];
      a[1] = xin[xbase + ((size_t)(c + 1) << 14) + (size_t)t * tmul] * dwm[kk * 2 + 1] + dwb[kk * 2 + 1];
#pragma unroll
      for (int j = 0; j < 8; ++j) {
        v2f b;
        b[0] = sWih[(j * 16 + lnr) * 32 + c];
        b[1] = sWih[(j * 16 + lnr) * 32 + c + 1];
        acc[j] = wmma4(a, b, acc[j]);
      }
    }
    // gates += h @ Whh^T
#pragma unroll
    for (int kk = 0; kk < 8; ++kk) {
      const int c = kk * 4 + khalf;
      v2f a;
      a[0] = ha[2 * kk];
      a[1] = ha[2 * kk + 1];
#pragma unroll
      for (int j = 0; j < 8; ++j) {
        v2f b;
        b[0] = sWhh[(j * 16 + lnr) * 32 + c];
        b[1] = sWhh[(j * 16 + lnr) * 32 + c + 1];
        acc[j] = wmma4(a, b, acc[j]);
      }
    }

    // i,f,g,o gate math; update c,h; write h to LDS (for A-relayout) + global
#pragma unroll
    for (int jj = 0; jj < 2; ++jj) {
#pragma unroll
      for (int r = 0; r < 8; ++r) {
        float ig = sigm(acc[jj][r] + bcol[jj]);
        float fg = sigm(acc[2 + jj][r] + bcol[2 + jj]);
        float gg = tanhf(acc[4 + jj][r] + bcol[4 + jj]);
        float og = sigm(acc[6 + jj][r] + bcol[6 + jj]);
        float cn = fg * cst[jj][r] + ig * gg;
        float hn = og * tanhf(cn);
        cst[jj][r] = cn;
        int lrow = rbase_off + r;
        shw[lrow * 32 + jj * 16 + lnr] = hn;
        int grow = m0 + lrow;
        hs[((size_t)grow << 12) + (t << 5) + jj * 16 + lnr] = hn;
      }
    }
    // same-wave LDS RAW: DS ops are in-order within a wave; compiler inserts
    // the dscnt wait before first use of ha.
#pragma unroll
    for (int kk = 0; kk < 8; ++kk) {
      ha[2 * kk]     = shw[lnr * 32 + kk * 4 + khalf];
      ha[2 * kk + 1] = shw[lnr * 32 + kk * 4 + khalf + 1];
    }
  }
}

// ---------------------------------------------------------------------------
// 1x1 conv (32->32) from hs (seq,T,32) layout into the concat buffer
// cat (4, 128, 128, 128) at channel offset c0off. GEMM: M=65536, N=32, K=32.
// Identical index formula for vertical and horizontal branches (spatial
// transpose of to_nchw_h is absorbed by the seq/t decomposition).
// ---------------------------------------------------------------------------
__global__ __launch_bounds__(256)
void conv1x1_seq_wmma(const float* __restrict__ hs, const float* __restrict__ wgt,
                      const float* __restrict__ bias, float* __restrict__ cat,
                      int c0off) {
  const int lane  = threadIdx.x & 31;
  const int wave  = threadIdx.x >> 5;
  const int lnr   = lane & 15;
  const int khalf = (lane >> 4) << 1;
  const int m0    = (blockIdx.x * 8 + wave) << 4;
  const int row   = m0 + lnr;

  v8f acc0 = v8zero(), acc1 = v8zero();
#pragma unroll
  for (int kk = 0; kk < 8; ++kk) {
    int k = kk * 4 + khalf;
    v2f a, b0, b1;
    a[0]  = hs[row * 32 + k];
    a[1]  = hs[row * 32 + k + 1];
    b0[0] = wgt[lnr * 32 + k];
    b0[1] = wgt[lnr * 32 + k + 1];
    b1[0] = wgt[(lnr + 16) * 32 + k];
    b1[1] = wgt[(lnr + 16) * 32 + k + 1];
    acc0 = wmma4(a, b0, acc0);
    acc1 = wmma4(a, b1, acc1);
  }

  float bb0 = bias[lnr], bb1 = bias[lnr + 16];
  const int rbase = m0 + ((lane >> 4) << 3);
#pragma unroll
  for (int r = 0; r < 8; ++r) {
    int m = rbase + r;
    int s = m >> 7, t = m & 127;
    int n = s >> 7, p = s & 127;
    size_t o0 = ((((size_t)n << 7) + (size_t)(c0off + lnr)) << 14) + (p << 7) + t;
    size_t o1 = ((((size_t)n << 7) + (size_t)(c0off + lnr + 16)) << 14) + (p << 7) + t;
    cat[o0] = acc0[r] + bb0;
    cat[o1] = acc1[r] + bb1;
  }
}

// ---------------------------------------------------------------------------
// Final 1x1 conv 32->1 + sigmoid. Tiny N dimension -> scalar VALU kernel.
// ---------------------------------------------------------------------------
__global__ __launch_bounds__(256)
void conv_out_sig(const float* __restrict__ in, const float* __restrict__ w,
                  float* __restrict__ out) {
  int idx = blockIdx.x * 256 + threadIdx.x;  // 0..65535
  int n = idx >> 14, hw = idx & 16383;
  const float* pin = in + (((size_t)n * 32) << 14) + hw;
  float s = 0.0f;
#pragma unroll
  for (int c = 0; c < 32; ++c) s += pin[(size_t)c << 14] * w[c];
  out[idx] = 1.0f / (1.0f + __expf(-s));
}

// ---------------------------------------------------------------------------
// Orchestration. Input index map (setup_inputs dict order, flattened):
//  0:x  1-6:attn(dead)  7:conv_in_w  8:conv2_w  9:conv3_w  10:conv_out_w
//  irnn1: 11 fm_w, 12 fm_b, 13-20 v-lstm(Wih,Whh,bih,bhh,Wih_r,Whh_r,bih_r,bhh_r),
//         21-28 h-lstm, 29 down_w, 30 down_b, 31 up_w, 32 up_b,
//         33 right_w, 34 right_b, 35 left_w, 36 left_b
//  irnn2: 37..62 (same structure)
// Concat channel order (irnn returns d,r,u,l): down->0, right->32, up->64, left->96.
// Workspace: A (conv activations, 8MB) | S (hs, 8MB) | CAT (32MB) = 48MB.
// ---------------------------------------------------------------------------
extern "C" void kernel_launch(void* const* d_in, const int* in_sizes, int n_in,
                              void* d_out, int out_size, void* d_ws, size_t ws_size,
                              hipStream_t stream) {
  (void)in_sizes; (void)n_in; (void)out_size; (void)ws_size;

  const float* x          = (const float*)d_in[0];
  const float* conv_in_w  = (const float*)d_in[7];
  const float* conv2_w    = (const float*)d_in[8];
  const float* conv3_w    = (const float*)d_in[9];
  const float* conv_out_w = (const float*)d_in[10];

  float* A   = (float*)d_ws;                       // 4*32*128*128  = 2,097,152 f
  float* S   = A + (size_t)2097152;                // 512*128*32    = 2,097,152 f
  float* CAT = S + (size_t)2097152;                // 4*128*128*128 = 8,388,608 f

  auto run_irnn = [&](int base, const float* src) {
    const float* fm_w   = (const float*)d_in[base + 0];
    const float* fm_b   = (const float*)d_in[base + 1];
    const float* vWih   = (const float*)d_in[base + 2];
    const float* vWhh   = (const float*)d_in[base + 3];
    const float* vbih   = (const float*)d_in[base + 4];
    const float* vbhh   = (const float*)d_in[base + 5];
    const float* vWih_r = (const float*)d_in[base + 6];
    const float* vWhh_r = (const float*)d_in[base + 7];
    const float* vbih_r = (const float*)d_in[base + 8];
    const float* vbhh_r = (const float*)d_in[base + 9];
    const float* hWih   = (const float*)d_in[base + 10];
    const float* hWhh   = (const float*)d_in[base + 11];
    const float* hbih   = (const float*)d_in[base + 12];
    const float* hbhh   = (const float*)d_in[base + 13];
    const float* hWih_r = (const float*)d_in[base + 14];
    const float* hWhh_r = (const float*)d_in[base + 15];
    const float* hbih_r = (const float*)d_in[base + 16];
    const float* hbhh_r = (const float*)d_in[base + 17];
    const float* down_w  = (const float*)d_in[base + 18];
    const float* down_b  = (const float*)d_in[base + 19];
    const float* up_w    = (const float*)d_in[base + 20];
    const float* up_b    = (const float*)d_in[base + 21];
    const float* right_w = (const float*)d_in[base + 22];
    const float* right_b = (const float*)d_in[base + 23];
    const float* left_w  = (const float*)d_in[base + 24];
    const float* left_b  = (const float*)d_in[base + 25];

    // vertical forward -> x_up (channels 64..95)
    lstm_scan_wmma<<<8, 128, 0, stream>>>(src, fm_w, fm_b, vWih, vWhh, vbih, vbhh, S, 1, 0);
    conv1x1_seq_wmma<<<512, 256, 0, stream>>>(S, up_w, up_b, CAT, 64);
    // vertical backward -> x_down (channels 0..31)
    lstm_scan_wmma<<<8, 128, 0, stream>>>(src, fm_w, fm_b, vWih_r, vWhh_r, vbih_r, vbhh_r, S, 1, 1);
    conv1x1_seq_wmma<<<512, 256, 0, stream>>>(S, down_w, down_b, CAT, 0);
    // horizontal forward -> x_right (channels 32..63)
    lstm_scan_wmma<<<8, 128, 0, stream>>>(src, fm_w, fm_b, hWih, hWhh, hbih, hbhh, S, 0, 0);
    conv1x1_seq_wmma<<<512, 256, 0, stream>>>(S, right_w, right_b, CAT, 32);
    // horizontal backward -> x_left (channels 96..127)
    lstm_scan_wmma<<<8, 128, 0, stream>>>(src, fm_w, fm_b, hWih_r, hWhh_r, hbih_r, hbhh_r, S, 0, 1);
    conv1x1_seq_wmma<<<512, 256, 0, stream>>>(S, left_w, left_b, CAT, 96);
  };

  // conv_in (attention branch is dead code in the reference -> skipped)
  conv3x3_wmma<<<512, 256, 0, stream>>>(x, conv_in_w, A, 32, 0);
  run_irnn(11, A);
  conv3x3_wmma<<<512, 256, 0, stream>>>(CAT, conv2_w, A, 128, 0);
  run_irnn(37, A);
  conv3x3_wmma<<<512, 256, 0, stream>>>(CAT, conv3_w, A, 128, 1);
  conv_out_sig<<<256, 256, 0, stream>>>(A, conv_out_w, (float*)d_out);
}